// LSAttention_87754771792310
// MI455X (gfx1250) — compile-verified
//
#include <hip/hip_runtime.h>
#include <hip/hip_bf16.h>
#include <math.h>

// Problem constants
#define B_    64
#define T_    1024
#define ENC_  512
#define RNN_  1024
#define ATT_  128
#define NF_   32
#define KER_  31
#define HALO_ 15

typedef __attribute__((ext_vector_type(2))) float v2f;
typedef __attribute__((ext_vector_type(8))) float v8f;

// Branch-free tanh: tanh(x) = 1 - 2/(exp(2x)+1); native v_exp_f32 + v_rcp_f32.
// Saturates correctly: x->+inf => exp2->inf => rcp->0 => 1; x->-inf => exp2->0 => -1.
__device__ __forceinline__ float fast_tanh(float x) {
    float e = __builtin_amdgcn_exp2f(x * 2.88539008177792681f); // 2*log2(e)
    return 1.0f - 2.0f * __builtin_amdgcn_rcpf(e + 1.0f);
}

// ---------------------------------------------------------------------------
// Kernel 1: one block per batch row. Fused conv -> (loc @ Wloc via WMMA f32
// 16x16x4) -> +q +W_memory -> tanh -> @We -> masked softmax -> w.
// 256 threads = 8 wave32s. All intermediates live in LDS.
// ---------------------------------------------------------------------------
__global__ __launch_bounds__(256)
void lsa_energy_softmax_kernel(const float* __restrict__ att_hs,
                               const float* __restrict__ W_memory,
                               const float* __restrict__ att_w,
                               const float* __restrict__ cum_w,
                               const unsigned char* __restrict__ mask,
                               const float* __restrict__ Wq,
                               const float* __restrict__ conv_k,
                               const float* __restrict__ Wloc,
                               const float* __restrict__ We,
                               float* __restrict__ w_out) {
    __shared__ float s_att[T_ + 2 * HALO_];     // padded att_weights (zero halo)
    __shared__ float s_cum[T_ + 2 * HALO_];     // padded cum_att_weights
    __shared__ float s_kern[KER_ * 2 * NF_];    // conv kernel [k][c][f]
    __shared__ float s_wloc[NF_ * ATT_];        // Wloc [32][128]
    __shared__ float s_q[ATT_];                 // q = att_hs @ Wq
    __shared__ float s_we[ATT_];                // We
    __shared__ float s_hs[RNN_];                // att_hs row
    __shared__ float s_feat[8][16 * NF_];       // per-wave 16x32 conv feature tile
    __shared__ float s_e[T_];                   // q-partials, then energies/softmax
    __shared__ float s_red[9];                  // block reduction

    const int b    = blockIdx.x;
    const int tid  = threadIdx.x;
    const int wv   = tid >> 5;
    const int lane = tid & 31;
    const int mrow = lane & 15;                 // M index within 16-lane half
    const int koff = (lane < 16) ? 0 : 2;       // K pair per A/B f32 layout
    const int roff = (lane < 16) ? 0 : 8;       // D-row offset per C/D layout

    // ---- stage inputs into LDS ----
    for (int i = tid; i < T_ + 2 * HALO_; i += 256) {
        int t = i - HALO_;
        bool in = (t >= 0) && (t < T_);
        s_att[i] = in ? att_w[b * T_ + t] : 0.0f;
        s_cum[i] = in ? cum_w[b * T_ + t] : 0.0f;
    }
    for (int i = tid; i < KER_ * 2 * NF_; i += 256) s_kern[i] = conv_k[i];
    for (int i = tid; i < NF_ * ATT_; i += 256)     s_wloc[i] = Wloc[i];
    for (int i = tid; i < ATT_; i += 256)           s_we[i]   = We[i];
    for (int i = tid; i < RNN_; i += 256)           s_hs[i]   = att_hs[b * RNN_ + i];
    __syncthreads();

    // ---- q = att_hs[b] @ Wq : 2-way split over RNN across all 256 threads ----
    {
        const int j = tid & 127;                // output column
        const int h = tid >> 7;                 // reduction half
        float acc = 0.0f;
        const int i0 = h * (RNN_ / 2);
        for (int i = i0; i < i0 + RNN_ / 2; ++i)
            acc += s_hs[i] * Wq[i * ATT_ + j];
        s_e[h * ATT_ + j] = acc;                // reuse s_e as q-partial scratch
    }
    __syncthreads();
    if (tid < ATT_) s_q[tid] = s_e[tid] + s_e[ATT_ + tid];
    __syncthreads();                            // s_e free again after this

    // ---- per-wave t-tiles: conv -> WMMA -> tanh-energy ----
    float* feat = &s_feat[wv][0];
    for (int mt = wv; mt < T_ / 16; mt += 8) {
        const int t0 = mt * 16;

        // conv features for this 16-row tile: lane == feature channel f
        for (int m = 0; m < 16; ++m) {
            float acc = 0.0f;
            const int base = t0 + m;            // s_att index already halo-shifted
            #pragma unroll
            for (int k = 0; k < KER_; ++k) {
                acc += s_att[base + k] * s_kern[k * 64 + lane];
                acc += s_cum[base + k] * s_kern[k * 64 + 32 + lane];
            }
            feat[m * NF_ + lane] = acc;
        }
        // same-wave LDS ops are in-order: safe to read back immediately

        // A fragments (16x4 F32 layout): VGPR0=.x holds K=koff, VGPR1=.y K=koff+1
        v2f afrag[8];
        #pragma unroll
        for (int ks = 0; ks < 8; ++ks) {
            afrag[ks].x = feat[mrow * NF_ + ks * 4 + koff];
            afrag[ks].y = feat[mrow * NF_ + ks * 4 + koff + 1];
        }

        float psum[8];
        #pragma unroll
        for (int r = 0; r < 8; ++r) psum[r] = 0.0f;

        for (int nt = 0; nt < ATT_ / 16; ++nt) {
            v8f c = {0.f, 0.f, 0.f, 0.f, 0.f, 0.f, 0.f, 0.f};
            #pragma unroll
            for (int ks = 0; ks < 8; ++ks) {
                v2f bfrag;                       // B 4x16: lane holds column N
                bfrag.x = s_wloc[(ks * 4 + koff)     * ATT_ + nt * 16 + mrow];
                bfrag.y = s_wloc[(ks * 4 + koff + 1) * ATT_ + nt * 16 + mrow];
                c = __builtin_amdgcn_wmma_f32_16x16x4_f32(
                        false, afrag[ks], false, bfrag, (short)0, c, false, false);
            }
            // fused epilogue: tanh(q + loc + W_memory) * We, per D layout
            const int   col = nt * 16 + mrow;
            const float qc  = s_q[col];
            const float wec = s_we[col];
            const float* wm = W_memory + ((size_t)b * T_ + t0) * ATT_ + col;
            #pragma unroll
            for (int r = 0; r < 8; ++r) {
                float wmv = wm[(size_t)(r + roff) * ATT_];
                psum[r] += fast_tanh(qc + wmv + c[r]) * wec;
            }
        }

        // row-sum across the 16 lanes of each half (halves stay disjoint)
        #pragma unroll
        for (int r = 0; r < 8; ++r) {
            float v = psum[r];
            #pragma unroll
            for (int off = 1; off < 16; off <<= 1) v += __shfl_xor(v, off, 32);
            if (mrow == 0) s_e[t0 + r + roff] = v;   // lanes 0 and 16 write
        }
    }
    __syncthreads();

    // ---- masked softmax over T (native exp2) ----
    float lmax = -INFINITY;
    for (int t = tid; t < T_; t += 256) {
        float e = s_e[t];
        if (!mask[b * T_ + t]) e = -INFINITY;
        s_e[t] = e;
        lmax = fmaxf(lmax, e);
    }
    #pragma unroll
    for (int off = 1; off < 32; off <<= 1) lmax = fmaxf(lmax, __shfl_xor(lmax, off, 32));
    if (lane == 0) s_red[wv] = lmax;
    __syncthreads();
    if (tid == 0) {
        float m = s_red[0];
        for (int i = 1; i < 8; ++i) m = fmaxf(m, s_red[i]);
        s_red[8] = m;
    }
    __syncthreads();
    const float bmax = s_red[8];

    float lsum = 0.0f;
    for (int t = tid; t < T_; t += 256) {
        float ex = __builtin_amdgcn_exp2f((s_e[t] - bmax) * 1.44269504088896341f);
        s_e[t] = ex;
        lsum += ex;
    }
    #pragma unroll
    for (int off = 1; off < 32; off <<= 1) lsum += __shfl_xor(lsum, off, 32);
    if (lane == 0) s_red[wv] = lsum;
    __syncthreads();
    if (tid == 0) {
        float s = s_red[0];
        for (int i = 1; i < 8; ++i) s += s_red[i];
        s_red[8] = s;
    }
    __syncthreads();
    const float inv = 1.0f / s_red[8];

    for (int t = tid; t < T_; t += 256)
        w_out[b * T_ + t] = s_e[t] * inv;
}

// ---------------------------------------------------------------------------
// Kernel 2: ctx[b,:] = sum_t w[t] * memory[b,t,:]. Streams the 128 MB memory
// tensor (kept RT-cached: the 160 MB working set fits MI455X's 192 MB L2, so
// replayed launches hit L2 above HBM bandwidth). grid = (4 d-chunks, 64
// batches), 256 threads = 64 float2 columns x 4 t-quarters.
// ---------------------------------------------------------------------------
__global__ __launch_bounds__(256)
void lsa_context_kernel(const float* __restrict__ memory,
                        const float* __restrict__ w,
                        float* __restrict__ ctx) {
    __shared__ float  s_w[T_];
    __shared__ float2 s_acc[256];

    const int b     = blockIdx.y;
    const int chunk = blockIdx.x;               // 4 chunks of 128 cols
    const int tid   = threadIdx.x;
    const int j     = tid & 63;                 // float2 column within chunk
    const int part  = tid >> 6;                 // t-quarter 0..3
    const int d0    = chunk * 128 + j * 2;

    for (int t = tid; t < T_; t += 256) s_w[t] = w[b * T_ + t];
    __syncthreads();

    const float2* mp = (const float2*)(memory + (size_t)b * T_ * ENC_ + d0);
    float2 acc; acc.x = 0.0f; acc.y = 0.0f;
    const int tbeg = part * (T_ / 4), tend = tbeg + (T_ / 4);
    for (int t = tbeg; t < tend; ++t) {
        const float  wv = s_w[t];
        const float2 m  = mp[(size_t)t * (ENC_ / 2)];
        acc.x += wv * m.x;
        acc.y += wv * m.y;
    }
    s_acc[tid] = acc;
    __syncthreads();
    if (part == 0) {
        float2 r = s_acc[j];
        #pragma unroll
        for (int p = 1; p < 4; ++p) {
            float2 o = s_acc[p * 64 + j];
            r.x += o.x; r.y += o.y;
        }
        float* out = ctx + (size_t)b * ENC_ + d0;
        out[0] = r.x;
        out[1] = r.y;
    }
}

extern "C" void kernel_launch(void* const* d_in, const int* in_sizes, int n_in,
                              void* d_out, int out_size, void* d_ws, size_t ws_size,
                              hipStream_t stream) {
    const float*         att_hs   = (const float*)d_in[0];
    const float*         memory   = (const float*)d_in[1];
    const float*         W_memory = (const float*)d_in[2];
    const float*         att_w    = (const float*)d_in[3];
    const float*         cum_w    = (const float*)d_in[4];
    const unsigned char* mmask    = (const unsigned char*)d_in[5];
    const float*         Wq       = (const float*)d_in[6];
    const float*         conv_k   = (const float*)d_in[7];
    const float*         Wloc     = (const float*)d_in[8];
    const float*         We       = (const float*)d_in[9];

    float* out = (float*)d_out;
    float* ctx = out;                 // [B, ENC] first
    float* w   = out + B_ * ENC_;     // then [B, T]

    lsa_energy_softmax_kernel<<<B_, 256, 0, stream>>>(
        att_hs, W_memory, att_w, cum_w, mmask, Wq, conv_k, Wloc, We, w);

    lsa_context_kernel<<<dim3(4, B_), 256, 0, stream>>>(memory, w, ctx);
}